// FeaturesLinear_37778532336329
// MI455X (gfx1250) — compile-verified
//
#include <hip/hip_runtime.h>
#include <hip/hip_bf16.h>

// ---------------------------------------------------------------------------
// FeaturesLinear for MI455X (gfx1250, wave32).
//
// out[b, d] = sum_{h=0..49} user_W[fid[50b+h], d] * rating_W[ridx[50b+h], d]
//           + item_W[item_ids[b], d] + bias[d]
//
// Memory-bound (~430 MB moved vs 0.21 GFLOP). Strategy:
//   * b128 gathers of user_W rows (rows are 512B, coalesced across a wave)
//   * stage the elementwise product in LDS
//   * do the 50-way segment reduction on the matrix pipe with
//     V_WMMA_F32_16X16X4_F32 and an indicator A-matrix (segments are
//     contiguous, so A[m,k] = (item_row(k) == m) costs 1 compare per element)
// ---------------------------------------------------------------------------

typedef __attribute__((ext_vector_type(2))) float v2f;
typedef __attribute__((ext_vector_type(8))) float v8f;

#define DIM            128
#define HIST           50
#define ROWS_PER_BLOCK 16                       // output rows per workgroup
#define ITEMS_PER_BLK  (ROWS_PER_BLOCK * HIST)  // 800
#define CHUNK          64                       // history items staged per pass
#define PITCH          136                      // LDS row pitch in floats:
                                                // 2*PITCH % 64 == 16 -> the two
                                                // half-wave B reads use disjoint banks

__global__ __launch_bounds__(256) void features_linear_wmma(
    const int* __restrict__ feature_ids,   // [T]
    const float* __restrict__ ratings,     // [T]
    const int* __restrict__ item_ids,      // [B]
    const float* __restrict__ user_W,      // [NUM_ITEMS+1, 128]
    const float* __restrict__ rating_W,    // [10, 128]
    const float* __restrict__ item_W,      // [NUM_ITEMS, 128]
    const float* __restrict__ bias,        // [128]
    float* __restrict__ out)               // [B, 128]
{
    __shared__ float lds_w[CHUNK * PITCH]; // 34,816 B staged product tile

    const int tid     = threadIdx.x;
    const int lane    = tid & 31;
    const int wave    = tid >> 5;          // 8 waves -> 8 dim-tiles of 16
    const int rowBase = blockIdx.x * ROWS_PER_BLOCK;
    const int hBase   = rowBase * HIST;    // first history item of this block

    // A-matrix addressing for this lane (16x4 f32 A: lane = M, VGPR/half = K)
    const int M  = lane & 15;
    const int kh = (lane >> 4) * 2;        // K offset contributed by lane half
    const int dimBase = wave * 16;

    v8f c = {0.f, 0.f, 0.f, 0.f, 0.f, 0.f, 0.f, 0.f};

    for (int cb = 0; cb < ITEMS_PER_BLK + CHUNK - 1 - ((ITEMS_PER_BLK - 1) % CHUNK);
         cb += CHUNK) {
        // ---------------- load phase: gather + multiply into LDS -----------
        // 256 threads cover 64 rows x 128 floats: 4 threads/row, 32 floats each
        {
            const int r    = tid >> 2;          // chunk-local history row 0..63
            const int part = tid & 3;
            const int d0   = part * 32;
            const int hh   = cb + r;            // block-local history index
            float4* lp = (float4*)&lds_w[r * PITCH + d0];
            if (hh < ITEMS_PER_BLK) {
                const int   hg  = hBase + hh;
                const int   fid = feature_ids[hg];
                const float rt  = ratings[hg];
                const int ridx  = (int)((rt - 0.5f) * 2.0f + 0.5f); // exact in fp32
                const float4* up = (const float4*)(user_W   + fid  * DIM + d0);
                const float4* rp = (const float4*)(rating_W + ridx * DIM + d0);
#pragma unroll
                for (int j = 0; j < 8; ++j) {
                    float4 u = up[j];
                    float4 rr = rp[j];
                    lp[j] = make_float4(u.x * rr.x, u.y * rr.y,
                                        u.z * rr.z, u.w * rr.w);
                }
            } else {
                // zero-fill tail rows so B operands are well-defined (0 * 0)
                const float4 z = make_float4(0.f, 0.f, 0.f, 0.f);
#pragma unroll
                for (int j = 0; j < 8; ++j) lp[j] = z;
            }
        }
        __syncthreads();

        // ---------------- WMMA phase: indicator-A segment reduction --------
        // One wave per 16-dim tile; 16 steps of K=4 cover the 64-item chunk.
#pragma unroll
        for (int s = 0; s < 16; ++s) {
            const int k0  = 4 * s + kh;        // chunk-local K for A.x / B.x
            const int hh0 = cb + k0;
            const int hh1 = hh0 + 1;

            v2f a, b;
            // A[m,k] = 1 iff history item k belongs to output row m (rows are
            // contiguous runs of 50) and the item is in range.
            a.x = (hh0 < ITEMS_PER_BLK && M == (hh0 / HIST)) ? 1.0f : 0.0f;
            a.y = (hh1 < ITEMS_PER_BLK && M == (hh1 / HIST)) ? 1.0f : 0.0f;
            // B layout (4x16 f32): lane = N, VGPR selects K {kh, kh+1}
            b.x = lds_w[k0       * PITCH + dimBase + M];
            b.y = lds_w[(k0 + 1) * PITCH + dimBase + M];

            c = __builtin_amdgcn_wmma_f32_16x16x4_f32(
                    /*neg_a=*/false, a, /*neg_b=*/false, b,
                    /*c_mod=*/(short)0, c, /*reuse_a=*/false, /*reuse_b=*/false);
        }
        __syncthreads();  // LDS reused next chunk
    }

    // ---------------- epilogue: + item_W gather + bias, store --------------
    // C/D layout: VGPR j -> (M=j, lanes 0-15) / (M=j+8, lanes 16-31), N=lane%16
    {
        const int N  = lane & 15;
        const int mh = (lane >> 4) * 8;
        const int dim = dimBase + N;
        const float bv = bias[dim];
#pragma unroll
        for (int j = 0; j < 8; ++j) {
            const int row = rowBase + mh + j;
            const int iid = item_ids[row];
            out[row * DIM + dim] = c[j] + item_W[iid * DIM + dim] + bv;
        }
    }
}

extern "C" void kernel_launch(void* const* d_in, const int* in_sizes, int n_in,
                              void* d_out, int out_size, void* d_ws, size_t ws_size,
                              hipStream_t stream) {
    const int*   feature_ids = (const int*)d_in[0];
    const float* ratings     = (const float*)d_in[1];
    // d_in[2] = segment_ids: unused — segments are contiguous (repeat(arange(B), 50))
    const int*   item_ids    = (const int*)d_in[3];
    const float* user_W      = (const float*)d_in[4];
    const float* rating_W    = (const float*)d_in[5];
    const float* item_W      = (const float*)d_in[6];
    const float* bias        = (const float*)d_in[7];
    float*       out         = (float*)d_out;

    const int batch  = in_sizes[3];                 // 16384
    const int blocks = batch / ROWS_PER_BLOCK;      // 1024

    features_linear_wmma<<<blocks, 256, 0, stream>>>(
        feature_ids, ratings, item_ids, user_W, rating_W, item_W, bias, out);
}